// GlobalBlock_16449724745524
// MI455X (gfx1250) — compile-verified
//
#include <hip/hip_runtime.h>

typedef __attribute__((ext_vector_type(2))) float v2f;
typedef __attribute__((ext_vector_type(8))) float v8f;

#define NB 256          // number of graphs / batches
#define D 128           // feature dim
#define CHUNK 1024      // rows per block in the scatter kernel

// ---------------------------------------------------------------------------
// Zero the scratch accumulators (sums + counts). d_ws is poisoned by the
// harness and must be re-initialized deterministically every call.
// ---------------------------------------------------------------------------
__global__ void zero_ws_kernel(float* __restrict__ ws, int n) {
  int i = blockIdx.x * blockDim.x + threadIdx.x;
  int stride = gridDim.x * blockDim.x;
  for (; i < n; i += stride) ws[i] = 0.0f;
}

// ---------------------------------------------------------------------------
// Masked segment-sum with sorted indices. Each wave32 streams one sub-row
// sequence with float4 (global_load_b128) loads, accumulates in registers,
// and flushes with global_atomic_add_f32 only on segment-index change.
// sums: [NB, D], counts: [NB] (count = number of rows per segment).
// ---------------------------------------------------------------------------
__global__ __launch_bounds__(256) void seg_sum_kernel(
    const float* __restrict__ x, const float* __restrict__ mask,
    const int* __restrict__ idx, int n_rows,
    float* __restrict__ sums, float* __restrict__ counts)
{
  const int tid  = threadIdx.x;
  const int lane = tid & 31;      // 32 lanes x float4 = 128 columns
  const int sub  = tid >> 5;      // 8 sub-rows per block iteration
  const int col4 = lane * 4;
  const float4* __restrict__ x4 = reinterpret_cast<const float4*>(x);

  const int rowBase = blockIdx.x * CHUNK;
  const int rowEnd  = min(rowBase + CHUNK, n_rows);

  float4 acc = make_float4(0.0f, 0.0f, 0.0f, 0.0f);
  float  cnt = 0.0f;
  int    cur = -1;

  for (int r = rowBase + sub; r < rowEnd; r += 8) {
    const int b = idx[r];                 // wave-uniform (same row per wave)
    if (b != cur) {
      if (cur >= 0) {
        float* s = sums + (size_t)cur * D + col4;
        atomicAdd(s + 0, acc.x); atomicAdd(s + 1, acc.y);
        atomicAdd(s + 2, acc.z); atomicAdd(s + 3, acc.w);
        if (lane == 0) atomicAdd(counts + cur, cnt);
      }
      acc = make_float4(0.0f, 0.0f, 0.0f, 0.0f);
      cnt = 0.0f;
      cur = b;
    }
    const float  m = mask[r];
    const float4 v = x4[(size_t)r * (D / 4) + lane];
    acc.x += v.x * m; acc.y += v.y * m;
    acc.z += v.z * m; acc.w += v.w * m;
    cnt += 1.0f;
  }
  if (cur >= 0) {
    float* s = sums + (size_t)cur * D + col4;
    atomicAdd(s + 0, acc.x); atomicAdd(s + 1, acc.y);
    atomicAdd(s + 2, acc.z); atomicAdd(s + 3, acc.w);
    if (lane == 0) atomicAdd(counts + cur, cnt);
  }
}

// ---------------------------------------------------------------------------
// Fused finalize + 2-layer MLP using V_WMMA_F32_16X16X4_F32 (full fp32).
// One wave32 per 16-row M tile (grid = 16 blocks, block = 32 threads).
//
// VGPR layouts per CDNA5 ISA 7.12.2:
//   A (16x4 f32, 2 VGPRs): lane m=L%16; lanes 0-15 hold K={0,1}, lanes 16-31
//     hold K={2,3}  -> a[j] = A[m][k + 2*(L>=16) + j]
//   B (4x16 f32, 2 VGPRs): row striped across lanes; VGPR j, lane half h:
//     K = j + 2*h, N = L%16 -> b[j] = B[k + 2*(L>=16) + j][n]
//   C/D (16x16 f32, 8 VGPRs): c[v] = D[v + 8*(L>=16)][L%16]
// ---------------------------------------------------------------------------
__global__ __launch_bounds__(32) void mlp_wmma_kernel(
    const float* __restrict__ global_attr,
    const float* __restrict__ sumsE, const float* __restrict__ cntE,
    const float* __restrict__ sumsN, const float* __restrict__ cntN,
    const float* __restrict__ W1, const float* __restrict__ b1,   // [384,32],[32]
    const float* __restrict__ W2, const float* __restrict__ b2,   // [32,128],[128]
    float* __restrict__ out)                                      // [256,128]
{
  __shared__ float A[16][384];   // collected tile (global | aggE | aggN)
  __shared__ float H[16][32];    // hidden tile after ReLU

  const int lane  = threadIdx.x;           // 0..31, whole wave active (EXEC=~0)
  const int row0  = blockIdx.x * 16;       // M tile base
  const int m     = lane & 15;
  const int n     = lane & 15;
  const int khalf = (lane >> 4) * 2;       // 0 or 2
  const int mhalf = (lane >> 4) * 8;       // 0 or 8

  // Stage collected[16][384] into LDS (with the scatter_mean division).
  for (int i = lane; i < 16 * 384; i += 32) {
    const int mr = i / 384;
    const int c  = i % 384;
    const int g  = row0 + mr;
    float v;
    if (c < 128) {
      v = global_attr[(size_t)g * D + c];
    } else if (c < 256) {
      v = sumsE[(size_t)g * D + (c - 128)] / fmaxf(cntE[g], 1.0f);
    } else {
      v = sumsN[(size_t)g * D + (c - 256)] / fmaxf(cntN[g], 1.0f);
    }
    A[mr][c] = v;
  }
  __syncthreads();

  // GEMM1: [16,384] @ [384,32] -> H, fused bias + ReLU.
  for (int nt = 0; nt < 2; ++nt) {
    v8f acc = {};
    for (int k = 0; k < 384; k += 4) {
      v2f af, bf;
      af.x = A[m][k + khalf];
      af.y = A[m][k + khalf + 1];
      bf.x = W1[(size_t)(k + khalf)     * 32 + nt * 16 + n];
      bf.y = W1[(size_t)(k + khalf + 1) * 32 + nt * 16 + n];
      acc = __builtin_amdgcn_wmma_f32_16x16x4_f32(
          false, af, false, bf, (short)0, acc, false, false);
    }
    const float bias = b1[nt * 16 + n];
#pragma unroll
    for (int v = 0; v < 8; ++v) {
      H[mhalf + v][nt * 16 + n] = fmaxf(acc[v] + bias, 0.0f);
    }
  }
  __syncthreads();

  // GEMM2: [16,32] @ [32,128] -> out, fused bias.
  for (int nt = 0; nt < 8; ++nt) {
    v8f acc = {};
    for (int k = 0; k < 32; k += 4) {
      v2f af, bf;
      af.x = H[m][k + khalf];
      af.y = H[m][k + khalf + 1];
      bf.x = W2[(size_t)(k + khalf)     * 128 + nt * 16 + n];
      bf.y = W2[(size_t)(k + khalf + 1) * 128 + nt * 16 + n];
      acc = __builtin_amdgcn_wmma_f32_16x16x4_f32(
          false, af, false, bf, (short)0, acc, false, false);
    }
    const float bias = b2[nt * 16 + n];
#pragma unroll
    for (int v = 0; v < 8; ++v) {
      out[(size_t)(row0 + mhalf + v) * 128 + nt * 16 + n] = acc[v] + bias;
    }
  }
}

// ---------------------------------------------------------------------------
// Launch
// ---------------------------------------------------------------------------
extern "C" void kernel_launch(void* const* d_in, const int* in_sizes, int n_in,
                              void* d_out, int out_size, void* d_ws, size_t ws_size,
                              hipStream_t stream) {
  const float* node_attr   = (const float*)d_in[0];
  const float* edge_attr   = (const float*)d_in[1];
  const float* global_attr = (const float*)d_in[2];
  const int*   ng_index    = (const int*)d_in[3];
  const int*   eg_index    = (const int*)d_in[4];
  const float* node_mask   = (const float*)d_in[5];
  const float* edge_mask   = (const float*)d_in[6];
  const float* W1          = (const float*)d_in[7];
  const float* b1          = (const float*)d_in[8];
  const float* W2          = (const float*)d_in[9];
  const float* b2          = (const float*)d_in[10];
  float*       out         = (float*)d_out;

  const int N = in_sizes[0] / D;   // 500000
  const int E = in_sizes[1] / D;   // 600000

  // Workspace layout: sumsE[NB*D] | sumsN[NB*D] | cntE[NB] | cntN[NB]
  float* ws    = (float*)d_ws;
  float* sumsE = ws;
  float* sumsN = ws + (size_t)NB * D;
  float* cntE  = ws + (size_t)2 * NB * D;
  float* cntN  = cntE + NB;
  const int ws_floats = 2 * NB * D + 2 * NB;

  zero_ws_kernel<<<64, 256, 0, stream>>>(ws, ws_floats);

  seg_sum_kernel<<<(E + CHUNK - 1) / CHUNK, 256, 0, stream>>>(
      edge_attr, edge_mask, eg_index, E, sumsE, cntE);
  seg_sum_kernel<<<(N + CHUNK - 1) / CHUNK, 256, 0, stream>>>(
      node_attr, node_mask, ng_index, N, sumsN, cntN);

  mlp_wmma_kernel<<<16, 32, 0, stream>>>(
      global_attr, sumsE, cntE, sumsN, cntN, W1, b1, W2, b2, out);
}